// LinearAttentionConditioner_4535485465062
// MI455X (gfx1250) — compile-verified
//
#include <hip/hip_runtime.h>
#include <hip/hip_bf16.h>
#include <math.h>

// MI455X / gfx1250: wave32, WMMA (V_WMMA_F32_16X16X4_F32) based chunked
// linear-attention scan. One wave per (batch, head); 16-wide chunks turn the
// sequential cumsum into 16x16 fp32 GEMMs on the matrix pipe.

typedef __attribute__((ext_vector_type(2))) float v2f;
typedef __attribute__((ext_vector_type(4))) float v4f;
typedef __attribute__((ext_vector_type(8))) float v8f;

#define N_    32
#define NH_   5
#define SLEN_ 3072
#define E_    16
#define T_    16
#define NCHUNK_ (SLEN_ / T_)

__device__ __forceinline__ float softplus_f(float x) {
  // numerically stable softplus, matches jax.nn.softplus in fp32
  return fmaxf(x, 0.0f) + log1pf(expf(-fabsf(x)));
}

__device__ __forceinline__ v8f wmma_f32(v2f a, v2f b, v8f c) {
  // D(16x16,f32) = A(16x4,f32) * B(4x16,f32) + C
  return __builtin_amdgcn_wmma_f32_16x16x4_f32(
      /*neg_a=*/false, a, /*neg_b=*/false, b,
      /*c_mod=*/(short)0, c, /*reuse_a=*/false, /*reuse_b=*/false);
}

// A-operand loader for a row-major tile M[row][col]: lane l<16 holds
// (row=l, col=k0..k0+1), lanes 16-31 hold (row=l, col=k0+2..k0+3).
__device__ __forceinline__ v2f ldP(const float* T, int kk, int stride) {
  int l = threadIdx.x & 15, hf = threadIdx.x >> 4;
  int k0 = kk * 4 + hf * 2;
  v2f r; r.x = T[l * stride + k0]; r.y = T[l * stride + k0 + 1]; return r;
}
// B-operand loader for a row-major tile M[krow][ncol]: lane l holds column l
// of rows k0 / k0+1 (upper lane half: rows k0+2 / k0+3).
__device__ __forceinline__ v2f ldQ(const float* T, int kk, int stride) {
  int l = threadIdx.x & 15, hf = threadIdx.x >> 4;
  int k0 = kk * 4 + hf * 2;
  v2f r; r.x = T[k0 * stride + l]; r.y = T[(k0 + 1) * stride + l]; return r;
}

__global__ __launch_bounds__(32)
void lin_attn_cond_kernel(const float* __restrict__ x,
                          const float* __restrict__ ew,
                          const float* __restrict__ eb,
                          const float* __restrict__ ind_in_w,
                          const float* __restrict__ ind_in_b,
                          const float* __restrict__ ind_out_w,
                          const float* __restrict__ ind_out_b,
                          const float* __restrict__ ind_res_w,
                          const float* __restrict__ fc1_w,
                          const float* __restrict__ fc1_b,
                          const float* __restrict__ fc2_w,
                          const float* __restrict__ fc2_b,
                          float* __restrict__ out) {
  __shared__ float Qs[T_ * E_], Ks[T_ * E_], Vs[T_ * E_];
  __shared__ float At[T_ * T_];        // masked Q K^T
  __shared__ float St[E_ * E_];        // running state S (row-major [i][j])
  __shared__ float Ps[T_ * E_];        // shift-by-one params tile (MLP input)
  __shared__ float Ht[T_ * 32];        // MLP hidden
  __shared__ float W1T[E_ * 32];       // fc1_w transposed: [e][o]
  __shared__ float W2T[32 * E_];       // fc2_w transposed: [o][p]
  __shared__ float B1s[32], B2s[16];
  __shared__ float zv[16], pvec[16], rsum[16], pscal[1];

  const int lane = threadIdx.x;
  const int l = lane & 15, hf = lane >> 4;
  const int nb = blockIdx.x / NH_;     // batch index
  const int hd = blockIdx.x % NH_;     // head index

  // ---------------- per-block init ----------------
  for (int i = lane; i < E_ * E_; i += 32) St[i] = 0.0f;
  if (lane < 16) zv[lane] = 0.0f;
  if (hd < 3) {
    for (int i = lane; i < 32 * 16; i += 32) {        // fc1_w (32,16) -> W1T[e][o]
      int o = i >> 4, e = i & 15;
      W1T[e * 32 + o] = fc1_w[i];
    }
    for (int i = lane; i < 16 * 32; i += 32) {        // fc2_w (16,32) -> W2T[o][p]
      int p = i >> 5, o = i & 31;
      W2T[o * 16 + p] = fc2_w[i];
    }
    B1s[lane] = fc1_b[lane];
    if (lane < 16) B2s[lane] = fc2_b[lane];
    if (lane < 16)
      pvec[lane] = (hd == 0) ? softplus_f(ind_in_w[lane])
                 : (hd == 1) ? ind_in_b[lane]
                             : ind_out_w[lane];
  } else if (lane == 0) {
    pscal[0] = (hd == 3) ? ind_out_b[0] : ind_res_w[0];
  }
  v8f Sacc = {};                                       // S in WMMA C-layout regs
  __syncthreads();

  const float* wq = ew + (size_t)hd * SLEN_ * E_;
  const float* wk = ew + (size_t)(hd + NH_) * SLEN_ * E_;
  const float* wv = ew + (size_t)(hd + 2 * NH_) * SLEN_ * E_;
  const float* bq = eb + (size_t)hd * SLEN_ * E_;
  const float* bk = eb + (size_t)(hd + NH_) * SLEN_ * E_;
  const float* bv = eb + (size_t)(hd + 2 * NH_) * SLEN_ * E_;
  const float* xb = x + (size_t)nb * SLEN_;

  const size_t SEC = (size_t)N_ * SLEN_ * E_;          // per-vector-output section
  const size_t sec_off = (hd < 3) ? (size_t)hd * SEC
                       : (hd == 3) ? 3 * SEC
                                   : 3 * SEC + (size_t)N_ * SLEN_;

  for (int c = 0; c < NCHUNK_; ++c) {
    const int s0 = c * T_;
    // ---------------- build Q,K,V tiles (softplus on q,k) ----------------
    const int f0 = lane * 8;                           // 8 contiguous elems/lane
    const float xv = xb[s0 + (lane >> 1)];             // one row of x per lane
    const size_t g = (size_t)s0 * E_ + f0;

    v4f a0 = *(const v4f*)(wq + g), a1 = *(const v4f*)(wq + g + 4);
    v4f c0 = *(const v4f*)(bq + g), c1 = *(const v4f*)(bq + g + 4);
    v4f q0, q1;
    #pragma unroll
    for (int j = 0; j < 4; ++j) { q0[j] = softplus_f(a0[j] * xv + c0[j]);
                                  q1[j] = softplus_f(a1[j] * xv + c1[j]); }
    *(v4f*)(Qs + f0) = q0; *(v4f*)(Qs + f0 + 4) = q1;

    a0 = *(const v4f*)(wk + g); a1 = *(const v4f*)(wk + g + 4);
    c0 = *(const v4f*)(bk + g); c1 = *(const v4f*)(bk + g + 4);
    #pragma unroll
    for (int j = 0; j < 4; ++j) { q0[j] = softplus_f(a0[j] * xv + c0[j]);
                                  q1[j] = softplus_f(a1[j] * xv + c1[j]); }
    *(v4f*)(Ks + f0) = q0; *(v4f*)(Ks + f0 + 4) = q1;

    a0 = *(const v4f*)(wv + g); a1 = *(const v4f*)(wv + g + 4);
    c0 = *(const v4f*)(bv + g); c1 = *(const v4f*)(bv + g + 4);
    #pragma unroll
    for (int j = 0; j < 4; ++j) { q0[j] = a0[j] * xv + c0[j];
                                  q1[j] = a1[j] * xv + c1[j]; }
    *(v4f*)(Vs + f0) = q0; *(v4f*)(Vs + f0 + 4) = q1;

    if (c + 1 < NCHUNK_) {                             // hide next chunk's HBM latency
      __builtin_prefetch(wq + g + T_ * E_, 0, 0);
      __builtin_prefetch(wk + g + T_ * E_, 0, 0);
      __builtin_prefetch(wv + g + T_ * E_, 0, 0);
      __builtin_prefetch(bq + g + T_ * E_, 0, 0);
      __builtin_prefetch(bk + g + T_ * E_, 0, 0);
      __builtin_prefetch(bv + g + T_ * E_, 0, 0);
    }
    __syncthreads();

    // ---------------- A = tril(Q K^T), inclusive diagonal ----------------
    v8f Aacc = {};
    #pragma unroll
    for (int kk = 0; kk < 4; ++kk)   // B = K^T  => lane-row pattern on Ks
      Aacc = wmma_f32(ldP(Qs, kk, 16), ldP(Ks, kk, 16), Aacc);
    #pragma unroll
    for (int r = 0; r < 8; ++r) {
      int M = r + 8 * hf;                              // C layout: (M, N=l)
      At[M * 16 + l] = (l <= M) ? Aacc[r] : 0.0f;
    }
    __syncthreads();

    // ------- numer = Q*S_prev + A*V ; denom = A*1 + Q*z_prev -------
    v8f Nacc = {}, Dacc = {};
    #pragma unroll
    for (int kk = 0; kk < 4; ++kk) {
      Nacc = wmma_f32(ldP(Qs, kk, 16), ldQ(St, kk, 16), Nacc);
      Nacc = wmma_f32(ldP(At, kk, 16), ldQ(Vs, kk, 16), Nacc);
      v2f ones = {1.0f, 1.0f};
      Dacc = wmma_f32(ldP(At, kk, 16), ones, Dacc);
      int k0 = kk * 4 + hf * 2;
      v2f zb; zb.x = zv[k0]; zb.y = zv[k0 + 1];        // z broadcast across N
      Dacc = wmma_f32(ldP(Qs, kk, 16), zb, Dacc);
    }
    float prm[8];
    #pragma unroll
    for (int r = 0; r < 8; ++r) prm[r] = Nacc[r] / Dacc[r];

    // ---------------- state update: S += K^T V ; z += sum(K) ----------------
    #pragma unroll
    for (int kk = 0; kk < 4; ++kk)   // A = K^T => k-row pattern on Ks
      Sacc = wmma_f32(ldQ(Ks, kk, 16), ldQ(Vs, kk, 16), Sacc);
    #pragma unroll
    for (int r = 0; r < 8; ++r) St[(r + 8 * hf) * 16 + l] = Sacc[r];
    if (lane < 16) {
      float zs = 0.0f;
      #pragma unroll
      for (int t = 0; t < 16; ++t) zs += Ks[t * 16 + lane];
      zv[lane] += zs;
    }
    __syncthreads();

    if (hd < 3) {
      // ------- shift-by-one: Ps row0 = carry, rows 1..15 = prm rows 0..14 -------
      float pv = (lane < 16) ? pvec[lane] : 0.0f;
      __syncthreads();
      #pragma unroll
      for (int r = 0; r < 8; ++r) {
        int M = r + 8 * hf;
        if (M < 15) Ps[(M + 1) * 16 + l] = prm[r];
        else        pvec[l] = prm[r];                  // carry row 15 -> next chunk
      }
      if (lane < 16) Ps[l] = pv;
      __syncthreads();

      // ------- MLP: hid = relu(Ps @ fc1_w^T + b1) -------
      v8f H0 = {}, H1 = {};
      #pragma unroll
      for (int kk = 0; kk < 4; ++kk) {
        v2f a = ldP(Ps, kk, 16);
        H0 = wmma_f32(a, ldQ(W1T, kk, 32), H0);        // out cols 0..15
        H1 = wmma_f32(a, ldQ(W1T + 16, kk, 32), H1);   // out cols 16..31
      }
      #pragma unroll
      for (int r = 0; r < 8; ++r) {
        int M = r + 8 * hf;
        Ht[M * 32 + l]      = fmaxf(H0[r] + B1s[l], 0.0f);
        Ht[M * 32 + 16 + l] = fmaxf(H1[r] + B1s[16 + l], 0.0f);
      }
      __syncthreads();

      // ------- o = hid @ fc2_w^T + b2 (K = 32 -> 8 wmma) -------
      v8f Oacc = {};
      #pragma unroll
      for (int kk = 0; kk < 8; ++kk)
        Oacc = wmma_f32(ldP(Ht, kk, 32), ldQ(W2T, kk, 16), Oacc);
      #pragma unroll
      for (int r = 0; r < 8; ++r) {
        int M = r + 8 * hf;
        float v = Oacc[r] + B2s[l];
        if (hd == 0) v = softplus_f(v);                // in_proj_weight = sp(...)
        out[sec_off + ((size_t)(nb * SLEN_ + s0 + M)) * E_ + l] = v;
      }
      __syncthreads();
    } else {
      // ------- heads 3,4: shifted row-sums over E -------
      if (lane == 0) rsum[0] = 0.0f;                   // keep LDS defined
      #pragma unroll
      for (int r = 0; r < 8; ++r) {
        float v = prm[r];
        v += __shfl_xor(v, 1, 32);
        v += __shfl_xor(v, 2, 32);
        v += __shfl_xor(v, 4, 32);
        v += __shfl_xor(v, 8, 32);                     // sum over 16-lane half
        if (l == 0) rsum[r + 8 * hf] = v;
      }
      __syncthreads();
      float prevS = pscal[0];
      __syncthreads();
      if (lane < 16) {
        float v = (lane == 0) ? prevS : rsum[lane - 1];
        out[sec_off + (size_t)(nb * SLEN_ + s0 + lane)] = v;
      }
      if (lane == 0) pscal[0] = rsum[15];
      __syncthreads();
    }
  }
}

extern "C" void kernel_launch(void* const* d_in, const int* in_sizes, int n_in,
                              void* d_out, int out_size, void* d_ws, size_t ws_size,
                              hipStream_t stream) {
  (void)in_sizes; (void)n_in; (void)out_size; (void)d_ws; (void)ws_size;
  const float* x         = (const float*)d_in[0];
  const float* ew        = (const float*)d_in[1];
  const float* eb        = (const float*)d_in[2];
  const float* ind_in_w  = (const float*)d_in[3];
  const float* ind_in_b  = (const float*)d_in[4];
  const float* ind_out_w = (const float*)d_in[5];
  const float* ind_out_b = (const float*)d_in[6];
  const float* ind_res_w = (const float*)d_in[7];
  const float* fc1_w     = (const float*)d_in[8];
  const float* fc1_b     = (const float*)d_in[9];
  const float* fc2_w     = (const float*)d_in[10];
  const float* fc2_b     = (const float*)d_in[11];
  float* out = (float*)d_out;

  dim3 grid(N_ * NH_);   // 160 blocks, one wave32 each: one (batch, head) pair
  dim3 block(32);
  lin_attn_cond_kernel<<<grid, block, 0, stream>>>(
      x, ew, eb, ind_in_w, ind_in_b, ind_out_w, ind_out_b, ind_res_w,
      fc1_w, fc1_b, fc2_w, fc2_b, out);
}